// DMPNNEncoder_2473901163248
// MI455X (gfx1250) — compile-verified
//
#include <hip/hip_runtime.h>

// ---------------- problem constants (from reference) ----------------
#define HIDDEN   300
#define NCOL     320                 // hidden padded to 20 tiles of 16
#define ATOM_DIM 133
#define BOND_DIM 14
#define MOL_FEAT 200
#define N_MOLS   4096
#define N_ATOMS  (N_MOLS * 32)       // 131072
#define N_BONDS  (2 * N_ATOMS)       // 262144
#define DEPTH    3
#define OUTW     (HIDDEN + MOL_FEAT) // 500

#define KP_I 160   // 147 -> 160
#define KP_M 320   // 300 -> 320
#define KP_A 448   // 433 -> 448

// ---------------- GEMM tiling ----------------
#define ROWS      32                 // bond/atom rows per block
#define A_STRIDE  18                 // LDS stride for A tile (even -> 8B-aligned float2)
#define BT_STRIDE 20                 // LDS stride for transposed B (conflict-free reads)
#define CT_PER_WAVE 5                // 20 col tiles / 4 col groups

typedef __attribute__((ext_vector_type(2))) float v2f;
typedef __attribute__((ext_vector_type(8))) float v8f;

// ---------------- tiny utility kernels ----------------
__global__ void zero_kernel(float* __restrict__ p, long n) {
    long t = (long)blockIdx.x * blockDim.x + threadIdx.x;
    if (t < n) p[t] = 0.0f;
}

// src is [K x HIDDEN], dst is [Kp x NCOL] zero-padded
__global__ void pad_weight_kernel(const float* __restrict__ src, float* __restrict__ dst,
                                  int K, int Kp) {
    int t = blockIdx.x * blockDim.x + threadIdx.x;
    if (t >= Kp * NCOL) return;
    int k = t / NCOL;
    int c = t - k * NCOL;
    dst[t] = (k < K && c < HIDDEN) ? src[k * HIDDEN + c] : 0.0f;
}

__global__ void pad_bias_kernel(const float* __restrict__ src, float* __restrict__ dst) {
    int c = threadIdx.x;
    if (c < NCOL) dst[c] = (c < HIDDEN) ? src[c] : 0.0f;
}

// one block per row: out[idx[r]][c] += h[r][c]   (stride NCOL both sides)
__global__ void scatter_add_kernel(const float* __restrict__ h, const int* __restrict__ idx,
                                   float* __restrict__ out) {
    int r = blockIdx.x;
    int c = threadIdx.x;            // blockDim = NCOL
    atomicAdd(&out[(long)idx[r] * NCOL + c], h[(long)r * NCOL + c]);
}

// out[mol][0:300] = 0 ; out[mol][300:500] = molecule_features
__global__ void init_out_kernel(const float* __restrict__ molfeat, float* __restrict__ out) {
    int t = blockIdx.x * blockDim.x + threadIdx.x;
    if (t >= N_MOLS * OUTW) return;
    int m = t / OUTW;
    int c = t - m * OUTW;
    out[t] = (c < HIDDEN) ? 0.0f : molfeat[m * MOL_FEAT + (c - HIDDEN)];
}

// one block per atom, 300 threads: out[mol[a]][c] += h_v[a][c]
__global__ void mol_readout_kernel(const float* __restrict__ hv, const int* __restrict__ a2m,
                                   float* __restrict__ out) {
    int a = blockIdx.x;
    int c = threadIdx.x;            // blockDim = HIDDEN
    atomicAdd(&out[(long)a2m[a] * OUTW + c], hv[(long)a * NCOL + c]);
}

// ---------------- fused gather + WMMA GEMM + bias/residual + relu ----------------
// MODE 0: h0    = relu([bond14 | atom133[src]] @ Wp + bp)                A1=bond, A2=atom, idx1=src
// MODE 1: hnew  = relu(h0 + (incoming[src] - h[b2rev]) @ Wp + bp)       A1=incoming, A2=h, idx1=src, idx2=b2rev, resid=h0
// MODE 2: h_v   = relu([atom133 | m_v300] @ Wp + bp)                    A1=atom, A2=m_v
template <int MODE>
__global__ __launch_bounds__(256)
void gemm_wmma_kernel(const float* __restrict__ A1, const float* __restrict__ A2,
                      const int* __restrict__ idx1, const int* __restrict__ idx2,
                      const float* __restrict__ Wp, const float* __restrict__ bp,
                      const float* __restrict__ resid, float* __restrict__ Dout,
                      int Kp) {
    __shared__ alignas(16) float Al[ROWS * A_STRIDE];
    __shared__ alignas(16) float Bt[NCOL * BT_STRIDE];   // transposed: Bt[col][k] for 16-k chunk
    __shared__ int sI1[ROWS];
    __shared__ int sI2[ROWS];

    const int tid  = threadIdx.x;
    const int wave = tid >> 5;
    const int lane = tid & 31;
    const int half = lane >> 4;       // WMMA lane-half (selects K pair / row group)
    const int nl   = lane & 15;
    const int row0 = blockIdx.x * ROWS;
    const int rowTile  = wave >> 2;   // 0..1
    const int ct0      = (wave & 3) * CT_PER_WAVE;

    if (tid < ROWS) {
        int r = row0 + tid;
        if (MODE == 0)      { sI1[tid] = idx1[r]; }
        else if (MODE == 1) { sI1[tid] = idx1[r]; sI2[tid] = idx2[r]; }
        else                { sI1[tid] = r; }
    }
    __syncthreads();

    v8f acc[CT_PER_WAVE] = {};

    for (int kk = 0; kk < Kp; kk += 16) {
        // stage 16 x NCOL weight slice, transposed into Bt[col][k]
        for (int i = tid; i < 16 * NCOL; i += 256) {
            int kr = i / NCOL;
            int c  = i - kr * NCOL;   // consecutive tid -> consecutive c: coalesced global read
            Bt[c * BT_STRIDE + kr] = Wp[(long)(kk + kr) * NCOL + c];
        }
        // stage ROWS x 16 A tile (mode-specific gather)
        for (int i = tid; i < ROWS * 16; i += 256) {
            int r  = i >> 4;
            int kc = i & 15;
            int k  = kk + kc;
            float v;
            if (MODE == 0) {
                int e = row0 + r;
                if (k < BOND_DIM)                 v = A1[(long)e * BOND_DIM + k];
                else if (k < BOND_DIM + ATOM_DIM) v = A2[(long)sI1[r] * ATOM_DIM + (k - BOND_DIM)];
                else                              v = 0.0f;
            } else if (MODE == 1) {
                v = A1[(long)sI1[r] * NCOL + k] - A2[(long)sI2[r] * NCOL + k];
            } else {
                int a = row0 + r;
                if (k < ATOM_DIM)                 v = A1[(long)a * ATOM_DIM + k];
                else if (k < ATOM_DIM + HIDDEN)   v = A2[(long)a * NCOL + (k - ATOM_DIM)];
                else                              v = 0.0f;
            }
            Al[r * A_STRIDE + kc] = v;
        }
        __syncthreads();

        // 4 k-steps of V_WMMA_F32_16X16X4_F32 per staged chunk
        #pragma unroll
        for (int ks = 0; ks < 16; ks += 4) {
            // A 16x4 f32 layout: lanes 0-15 hold K={ks,ks+1}, lanes 16-31 K={ks+2,ks+3}
            v2f a = *(const v2f*)&Al[(rowTile * 16 + nl) * A_STRIDE + ks + 2 * half];
            #pragma unroll
            for (int i = 0; i < CT_PER_WAVE; i++) {
                int cb = (ct0 + i) * 16 + nl;
                // B 4x16 layout: per lane K pair {ks|ks+2, +1|+3} by half -> contiguous in Bt
                v2f b = *(const v2f*)&Bt[cb * BT_STRIDE + ks + 2 * half];
                acc[i] = __builtin_amdgcn_wmma_f32_16x16x4_f32(
                    false, a, false, b, (short)0, acc[i], false, false);
            }
        }
        __syncthreads();
    }

    // epilogue: C/D layout VGPR j -> row (j + 8*half), col nl
    const int rbase = row0 + rowTile * 16 + half * 8;
    #pragma unroll
    for (int i = 0; i < CT_PER_WAVE; i++) {
        int col = (ct0 + i) * 16 + nl;
        float bias = bp[col];
        #pragma unroll
        for (int j = 0; j < 8; j++) {
            long r = rbase + j;
            float v = acc[i][j] + bias;
            if (MODE == 1) v += resid[r * NCOL + col];
            Dout[r * NCOL + col] = v > 0.0f ? v : 0.0f;
        }
    }
}

// ---------------- orchestration ----------------
extern "C" void kernel_launch(void* const* d_in, const int* in_sizes, int n_in,
                              void* d_out, int out_size, void* d_ws, size_t ws_size,
                              hipStream_t stream) {
    (void)in_sizes; (void)n_in; (void)out_size; (void)ws_size;

    const float* atom_feat = (const float*)d_in[0];
    const float* bond_feat = (const float*)d_in[1];
    const int*   bidx      = (const int*)  d_in[2];
    const float* mol_feat  = (const float*)d_in[3];
    const int*   a2m       = (const int*)  d_in[4];
    const int*   b2rev     = (const int*)  d_in[5];
    const float* W_i       = (const float*)d_in[6];
    const float* b_i       = (const float*)d_in[7];
    const float* W_m       = (const float*)d_in[8];
    const float* b_m       = (const float*)d_in[9];
    const float* W_a       = (const float*)d_in[10];
    const float* b_a       = (const float*)d_in[11];
    float* out = (float*)d_out;

    const int* src = bidx;
    const int* dst = bidx + N_BONDS;

    // workspace carve-up (floats)
    float* p = (float*)d_ws;
    float* Wp_i = p;  p += (long)KP_I * NCOL;
    float* Wp_m = p;  p += (long)KP_M * NCOL;
    float* Wp_a = p;  p += (long)KP_A * NCOL;
    float* bp_i = p;  p += NCOL;
    float* bp_m = p;  p += NCOL;
    float* bp_a = p;  p += NCOL;
    float* h0   = p;  p += (long)N_BONDS * NCOL;   // also reused for h_v at the end
    float* hA   = p;  p += (long)N_BONDS * NCOL;
    float* hB   = p;  p += (long)N_BONDS * NCOL;
    float* incoming = p; p += (long)N_ATOMS * NCOL; // also reused as m_v

    // 1) pad weights/biases into WMMA-friendly zero-padded buffers
    pad_weight_kernel<<<(KP_I * NCOL + 255) / 256, 256, 0, stream>>>(W_i, Wp_i, BOND_DIM + ATOM_DIM, KP_I);
    pad_weight_kernel<<<(KP_M * NCOL + 255) / 256, 256, 0, stream>>>(W_m, Wp_m, HIDDEN, KP_M);
    pad_weight_kernel<<<(KP_A * NCOL + 255) / 256, 256, 0, stream>>>(W_a, Wp_a, ATOM_DIM + HIDDEN, KP_A);
    pad_bias_kernel<<<1, NCOL, 0, stream>>>(b_i, bp_i);
    pad_bias_kernel<<<1, NCOL, 0, stream>>>(b_m, bp_m);
    pad_bias_kernel<<<1, NCOL, 0, stream>>>(b_a, bp_a);

    // 2) h0 = relu([bond | atom[src]] @ W_i + b_i)
    gemm_wmma_kernel<0><<<N_BONDS / ROWS, 256, 0, stream>>>(
        bond_feat, atom_feat, src, nullptr, Wp_i, bp_i, nullptr, h0, KP_I);

    // 3) DEPTH message-passing iterations
    const long incN = (long)N_ATOMS * NCOL;
    const float* hcur = h0;
    float* hbuf[2] = {hA, hB};
    for (int d = 0; d < DEPTH; d++) {
        zero_kernel<<<(int)((incN + 255) / 256), 256, 0, stream>>>(incoming, incN);
        scatter_add_kernel<<<N_BONDS, NCOL, 0, stream>>>(hcur, dst, incoming);
        float* hnext = hbuf[d & 1];
        gemm_wmma_kernel<1><<<N_BONDS / ROWS, 256, 0, stream>>>(
            incoming, hcur, src, b2rev, Wp_m, bp_m, h0, hnext, KP_M);
        hcur = hnext;
    }

    // 4) m_v = segment_sum(h over src)   (reuse `incoming`)
    zero_kernel<<<(int)((incN + 255) / 256), 256, 0, stream>>>(incoming, incN);
    scatter_add_kernel<<<N_BONDS, NCOL, 0, stream>>>(hcur, src, incoming);

    // 5) h_v = relu([atom | m_v] @ W_a + b_a)   (reuse h0 buffer for h_v)
    float* h_v = h0;
    gemm_wmma_kernel<2><<<N_ATOMS / ROWS, 256, 0, stream>>>(
        atom_feat, incoming, nullptr, nullptr, Wp_a, bp_a, nullptr, h_v, KP_A);

    // 6) molecule readout + feature concat
    init_out_kernel<<<(N_MOLS * OUTW + 255) / 256, 256, 0, stream>>>(mol_feat, out);
    mol_readout_kernel<<<N_ATOMS, HIDDEN, 0, stream>>>(h_v, a2m, out);
}